// ConvBlock_60971355734041
// MI455X (gfx1250) — compile-verified
//
#include <hip/hip_runtime.h>

#define N_NODES 20000
#define N_EDGES 640000
#define D 128

typedef float v2f __attribute__((ext_vector_type(2)));
typedef float v8f __attribute__((ext_vector_type(8)));

// ---------------------------------------------------------------------------
// Scatter-add aggregation: agg[dst] += x[src] for each edge.
// 1 thread per (edge, 4-float chunk) -> E*32 threads, exactly 80000 blocks.
// float4 gather from x, 4x hardware f32 atomic adds into L2-resident agg.
// ---------------------------------------------------------------------------
__global__ __launch_bounds__(256) void scatter_add_kernel(
    const float* __restrict__ x, const long long* __restrict__ ei,
    float* __restrict__ agg) {
  unsigned t = blockIdx.x * 256u + threadIdx.x;  // < E*32 = 20.48M
  unsigned e = t >> 5;
  unsigned c = (t & 31u) << 2;  // float offset within the row (0,4,...,124)
  int src = (int)ei[e];
  int dst = (int)ei[N_EDGES + e];
  const float4 v = *reinterpret_cast<const float4*>(x + (size_t)src * D + c);
  float* p = agg + (size_t)dst * D + c;
  unsafeAtomicAdd(p + 0, v.x);
  unsafeAtomicAdd(p + 1, v.y);
  unsafeAtomicAdd(p + 2, v.z);
  unsafeAtomicAdd(p + 3, v.w);
}

// ---------------------------------------------------------------------------
// Fused GraphConv linear stage: out = [relu]( agg @ Wrel + brel + xin @ Wroot )
// One block = 16 node rows. 8 waves, each wave owns a 16x16 output tile and
// accumulates with v_wmma_f32_16x16x4_f32 over K=128 (32 WMMA per GEMM).
//
// f32 16x4 A-operand lane layout (ISA 7.12.2):
//   lane L: half = L/16, m = L%16
//   a[0] = A[m][k + 2*half], a[1] = A[m][k + 2*half + 1]
// B (4x16, row striped across lanes): b[0] = B[k+2*half][n], b[1] = B[k+2*half+1][n]
// C/D: VGPR r holds out[r + 8*half][n], n = lane%16.
// ---------------------------------------------------------------------------
template <bool RELU>
__global__ __launch_bounds__(256) void graphconv_gemm_kernel(
    const float* __restrict__ agg, const float* __restrict__ xin,
    const float* __restrict__ Wrel, const float* __restrict__ brel,
    const float* __restrict__ Wroot, float* __restrict__ out) {
  __shared__ float sA[16][132];  // agg tile, padded: stride 132 -> conflict-free column reads
  __shared__ float sX[16][132];  // x tile

  const int tid = threadIdx.x;
  const int row0 = blockIdx.x * 16;

  // Cooperative coalesced load of both 16x128 tiles (512 float4 each).
  for (int i = tid; i < 16 * 32; i += 256) {
    const int r = i >> 5;
    const int c4 = i & 31;
    reinterpret_cast<float4*>(&sA[r][0])[c4] =
        reinterpret_cast<const float4*>(agg + (size_t)(row0 + r) * D)[c4];
    reinterpret_cast<float4*>(&sX[r][0])[c4] =
        reinterpret_cast<const float4*>(xin + (size_t)(row0 + r) * D)[c4];
  }
  __syncthreads();

  const int wave = tid >> 5;   // 0..7 -> which 16-column slice of the output
  const int lane = tid & 31;
  const int half = lane >> 4;  // 0/1
  const int m = lane & 15;     // row within tile (A) / column within tile (B,C,D)
  const int ncol = wave * 16 + m;

  v8f acc = {};

#pragma unroll
  for (int k = 0; k < D; k += 4) {
    v2f a, b;
    a.x = sA[m][k + 2 * half];
    a.y = sA[m][k + 2 * half + 1];
    b.x = Wrel[(size_t)(k + 2 * half) * D + ncol];
    b.y = Wrel[(size_t)(k + 2 * half + 1) * D + ncol];
    acc = __builtin_amdgcn_wmma_f32_16x16x4_f32(
        false, a, false, b, (short)0, acc, false, false);
  }

#pragma unroll
  for (int k = 0; k < D; k += 4) {
    v2f a, b;
    a.x = sX[m][k + 2 * half];
    a.y = sX[m][k + 2 * half + 1];
    b.x = Wroot[(size_t)(k + 2 * half) * D + ncol];
    b.y = Wroot[(size_t)(k + 2 * half + 1) * D + ncol];
    acc = __builtin_amdgcn_wmma_f32_16x16x4_f32(
        false, a, false, b, (short)0, acc, false, false);
  }

  const float bias = brel[ncol];
#pragma unroll
  for (int r = 0; r < 8; ++r) {
    float v = acc[r] + bias;
    if (RELU) v = fmaxf(v, 0.0f);
    out[(size_t)(row0 + r + 8 * half) * D + ncol] = v;
  }
}

// ---------------------------------------------------------------------------
// Launch: layer1 h (with ReLU) is written into d_out, then reused in-place by
// layer2 (each block reads its own 16 rows into LDS before overwriting them,
// and no block reads another block's rows -> race-free). ws holds one agg
// buffer (10.24 MB), zeroed before each scatter.
// ---------------------------------------------------------------------------
extern "C" void kernel_launch(void* const* d_in, const int* in_sizes, int n_in,
                              void* d_out, int out_size, void* d_ws, size_t ws_size,
                              hipStream_t stream) {
  const float* x = (const float*)d_in[0];
  const long long* ei = (const long long*)d_in[1];  // int64 edge_index [2, E]
  const float* W1_rel = (const float*)d_in[2];
  const float* b1_rel = (const float*)d_in[3];
  const float* W1_root = (const float*)d_in[4];
  const float* W2_rel = (const float*)d_in[5];
  const float* b2_rel = (const float*)d_in[6];
  const float* W2_root = (const float*)d_in[7];
  float* out = (float*)d_out;

  const size_t aggBytes = (size_t)N_NODES * D * sizeof(float);
  float* agg = (float*)d_ws;
  float* h = out;  // layer-1 activations live in d_out (safe, see above)

  const dim3 sblk(256);
  const dim3 sgrid((unsigned)((size_t)N_EDGES * 32 / 256));  // exact
  const dim3 ggrid(N_NODES / 16);                            // exact (1250)

  // ---- Layer 1 ----
  hipMemsetAsync(agg, 0, aggBytes, stream);
  scatter_add_kernel<<<sgrid, sblk, 0, stream>>>(x, ei, agg);
  graphconv_gemm_kernel<true><<<ggrid, sblk, 0, stream>>>(
      agg, x, W1_rel, b1_rel, W1_root, h);

  // ---- Layer 2 ----
  hipMemsetAsync(agg, 0, aggBytes, stream);
  scatter_add_kernel<<<sgrid, sblk, 0, stream>>>(h, ei, agg);
  graphconv_gemm_kernel<false><<<ggrid, sblk, 0, stream>>>(
      agg, h, W2_rel, b2_rel, W2_root, out);
}